// DeltaMHCBlock_89721866813916
// MI455X (gfx1250) — compile-verified
//
#include <hip/hip_runtime.h>
#include <hip/hip_bf16.h>
#include <cstdint>

typedef __attribute__((ext_vector_type(2))) float v2f;
typedef __attribute__((ext_vector_type(8))) float v8f;

#define WMMA_F32(a, b, c) \
  __builtin_amdgcn_wmma_f32_16x16x4_f32(false, (a), false, (b), (short)0, (c), false, false)

namespace {
constexpr int B = 2, S = 2048, D = 1024;
constexpr int NH = 16, HD = 64, NS = 8, SDIM = 128;
constexpr int D3 = 3 * D, D4 = 4 * D;
constexpr int ROWS = B * S;        // 4096
constexpr int CL = 32, CH = S / CL; // chunked scan: 64 chunks of 32
constexpr float EPS = 1e-6f;

// workspace layout (float offsets)
constexpr size_t OFF_XD  = 0;                                  // x_delta          [ROWS*D]
constexpr size_t OFF_N   = OFF_XD  + (size_t)ROWS * D;         // normed1/normed2  [ROWS*D]
constexpr size_t OFF_BIG = OFF_N   + (size_t)ROWS * D;         // qkv / mlp hidden [ROWS*D3]
constexpr size_t OFF_AO  = OFF_BIG + (size_t)ROWS * D3;        // attn_o -> ffn    [ROWS*D]
constexpr size_t OFF_Y1  = OFF_AO  + (size_t)ROWS * D;         // x_delta+attn_out [ROWS*D]
constexpr size_t OFF_CTR = OFF_Y1  + (size_t)ROWS * D;         // chunk contribs   [B*D*CH]
constexpr size_t OFF_SIN = OFF_CTR + (size_t)B * D * CH;       // chunk start state
constexpr size_t OFF_PS  = OFF_SIN + (size_t)B * D * CH;       // chunk out-sums
constexpr size_t OFF_XST = OFF_PS  + (size_t)B * D * CH;       // x_state [B*D]
constexpr size_t OFF_GR  = OFF_XST + (size_t)B * D;            // g_res   [B*NS]
constexpr size_t OFF_H   = OFF_GR  + (size_t)B * NS;           // H       [NS*NS]
} // namespace

__device__ __forceinline__ float sigmoidf_(float v) { return 1.0f / (1.0f + __expf(-v)); }

// ---------------- delta operator: 3-phase chunked scan ----------------
__global__ __launch_bounds__(256) void k_delta_a(const float* __restrict__ x,
                                                 const float* __restrict__ decay,
                                                 float* __restrict__ contrib) {
  int idx = blockIdx.x * 256 + threadIdx.x;   // (b, c, d) flat, B*D*CH
  int d = idx % D;
  int t = idx / D;
  int c = t % CH, b = t / CH;
  float beta = sigmoidf_(decay[d]);
  const float* xp = x + ((size_t)(b * S + c * CL)) * D + d;
  float s = 0.f;
#pragma unroll
  for (int i = 0; i < CL; ++i) {
    float xt = xp[(size_t)i * D];
    s = beta * s + (1.f - beta) * xt;
  }
  contrib[idx] = s;
}

__global__ __launch_bounds__(256) void k_delta_b(const float* __restrict__ contrib,
                                                 const float* __restrict__ decay,
                                                 float* __restrict__ sinb) {
  int i = blockIdx.x * 256 + threadIdx.x;
  if (i >= B * D) return;
  int d = i % D, b = i / D;
  float beta = sigmoidf_(decay[d]);
  float f = __powf(beta, (float)CL);
  float s = 0.f;
  for (int c = 0; c < CH; ++c) {
    size_t j = ((size_t)(b * CH + c)) * D + d;
    sinb[j] = s;
    s = f * s + contrib[j];
  }
}

__global__ __launch_bounds__(256) void k_delta_c(const float* __restrict__ x,
                                                 const float* __restrict__ decay,
                                                 const float* __restrict__ sinb,
                                                 float* __restrict__ xd,
                                                 float* __restrict__ psum) {
  int idx = blockIdx.x * 256 + threadIdx.x;
  int d = idx % D;
  int t = idx / D;
  int c = t % CH, b = t / CH;
  float beta = sigmoidf_(decay[d]);
  float s = sinb[idx];
  const float* xp = x + ((size_t)(b * S + c * CL)) * D + d;
  float* op = xd + ((size_t)(b * S + c * CL)) * D + d;
  float acc = 0.f;
#pragma unroll
  for (int i = 0; i < CL; ++i) {
    float xt = xp[(size_t)i * D];
    float o = xt - s;
    op[(size_t)i * D] = o;
    acc += o;
    s = beta * s + (1.f - beta) * xt;
  }
  psum[idx] = acc;
}

// ------------- x_state reduce + gates + sinkhorn (tiny, one block) -------------
__global__ __launch_bounds__(256) void k_combo(const float* __restrict__ psum,
                                               const float* __restrict__ gw,
                                               const float* __restrict__ gb,
                                               const float* __restrict__ phi,
                                               float* __restrict__ xstate,
                                               float* __restrict__ gres,
                                               float* __restrict__ Hm) {
  __shared__ float xs[B * D];
  for (int i = threadIdx.x; i < B * D; i += 256) {
    int b = i / D, d = i % D;
    const float* p = psum + (size_t)b * CH * D + d;
    float s = 0.f;
    for (int c = 0; c < CH; ++c) s += p[(size_t)c * D];
    s /= (float)S;
    xs[i] = s;
    xstate[i] = s;
  }
  __syncthreads();
  if (threadIdx.x < B * NS) {
    int b = threadIdx.x / NS;
    int o = 2 * NS + (threadIdx.x % NS);   // only gates[:, 2n:] are used
    float a = gb[o];
    const float* wr = gw + (size_t)o * D;
    const float* xr = xs + b * D;
    for (int d = 0; d < D; ++d) a += xr[d] * wr[d];
    gres[threadIdx.x] = sigmoidf_(a);
  }
  if (threadIdx.x == 0) {
    float Kk[NS][NS];
    for (int i = 0; i < NS; ++i)
      for (int j = 0; j < NS; ++j) Kk[i][j] = __expf(phi[i * NS + j]);
    for (int it = 0; it < 15; ++it) {
      for (int i = 0; i < NS; ++i) {
        float rs = 0.f;
        for (int j = 0; j < NS; ++j) rs += Kk[i][j];
        float inv = 1.f / rs;
        for (int j = 0; j < NS; ++j) Kk[i][j] *= inv;
      }
      for (int j = 0; j < NS; ++j) {
        float cs = 0.f;
        for (int i = 0; i < NS; ++i) cs += Kk[i][j];
        float inv = 1.f / cs;
        for (int i = 0; i < NS; ++i) Kk[i][j] *= inv;
      }
    }
    for (int i = 0; i < NS; ++i)
      for (int j = 0; j < NS; ++j) Hm[i * NS + j] = Kk[i][j];
  }
}

// ---------------- RMS norm: one row (D=1024) per 256-thread block ----------------
__global__ __launch_bounds__(256) void k_rms(const float* __restrict__ x,
                                             const float* __restrict__ w,
                                             float* __restrict__ out) {
  __shared__ float red[8];
  int row = blockIdx.x;
  const float* xr = x + (size_t)row * D;
  float v[4];
  float ss = 0.f;
#pragma unroll
  for (int j = 0; j < 4; ++j) {
    v[j] = xr[threadIdx.x + j * 256];
    ss += v[j] * v[j];
  }
#pragma unroll
  for (int off = 16; off > 0; off >>= 1) ss += __shfl_xor(ss, off, 32);
  int wave = threadIdx.x >> 5, lane = threadIdx.x & 31;
  if (lane == 0) red[wave] = ss;
  __syncthreads();
  float tot = 0.f;
#pragma unroll
  for (int i = 0; i < 8; ++i) tot += red[i];
  float r = rsqrtf(tot / (float)D + EPS);
#pragma unroll
  for (int j = 0; j < 4; ++j)
    out[(size_t)row * D + threadIdx.x + j * 256] = v[j] * r * w[threadIdx.x + j * 256];
}

// ---------------- generic WMMA f32 GEMM: C = act(A[M,K] . W[N,K]^T + bias) (+res) -------
// wave tile 32x32 (2x2 of 16x16x4), block = 8 waves -> 64x128 tile
template <int ACT, bool RES>
__global__ __launch_bounds__(256) void k_gemm(const float* __restrict__ A,
                                              const float* __restrict__ W,
                                              const float* __restrict__ bias,
                                              const float* __restrict__ res,
                                              float* __restrict__ C,
                                              int M, int N, int K) {
  const int lane = threadIdx.x & 31;
  const int wave = threadIdx.x >> 5;
  const int lr = lane & 15, lh = lane >> 4;
  const int mrow = blockIdx.x * 64 + (wave >> 2) * 32;
  const int ncol = blockIdx.y * 128 + (wave & 3) * 32;
  const float* a0 = A + (size_t)(mrow + lr) * K + lh * 2;
  const float* a1 = a0 + (size_t)16 * K;
  const float* b0 = W + (size_t)(ncol + lr) * K + lh * 2;
  const float* b1 = b0 + (size_t)16 * K;
  v8f c00 = {0, 0, 0, 0, 0, 0, 0, 0}, c01 = c00, c10 = c00, c11 = c00;
#pragma unroll 4
  for (int k = 0; k < K; k += 4) {
    v2f av0 = *(const v2f*)(a0 + k);
    v2f av1 = *(const v2f*)(a1 + k);
    v2f bv0 = *(const v2f*)(b0 + k);
    v2f bv1 = *(const v2f*)(b1 + k);
    c00 = WMMA_F32(av0, bv0, c00);
    c01 = WMMA_F32(av0, bv1, c01);
    c10 = WMMA_F32(av1, bv0, c10);
    c11 = WMMA_F32(av1, bv1, c11);
  }
  v8f accs[2][2] = {{c00, c01}, {c10, c11}};
#pragma unroll
  for (int mi = 0; mi < 2; ++mi)
#pragma unroll
    for (int ni = 0; ni < 2; ++ni) {
      int col = ncol + ni * 16 + lr;
      float bb = bias[col];
#pragma unroll
      for (int r = 0; r < 8; ++r) {
        int rowi = mrow + mi * 16 + lh * 8 + r;
        float vv = accs[mi][ni][r] + bb;
        if (ACT == 1) vv = 0.5f * vv * (1.0f + erff(vv * 0.70710678118654752f));
        size_t idx = (size_t)rowi * N + col;
        if (RES) vv += res[idx];
        C[idx] = vv;
      }
    }
}

// ---------------- flash attention (causal), one 16-query tile per wave ----------------
__global__ __launch_bounds__(128) void k_flash(const float* __restrict__ qkv,
                                               float* __restrict__ ao) {
  __shared__ float pl[4][16 * 18];
  const int lane = threadIdx.x & 31;
  const int wave = threadIdx.x >> 5;
  const int lr = lane & 15, lh = lane >> 4;
  const int qtile = blockIdx.x * 4 + wave;
  const int bh = blockIdx.y;
  const int b = bh >> 4, h = bh & 15;
  const int qbase = qtile * 16;
  float* wp = pl[wave];

  // preload Q fragments, pre-scaled by 1/sqrt(64)
  const float* qrow = qkv + ((size_t)(b * S + qbase + lr)) * D3 + h * HD + lh * 2;
  v2f qf[16];
#pragma unroll
  for (int dk = 0; dk < 16; ++dk) {
    v2f q = *(const v2f*)(qrow + dk * 4);
    q.x *= 0.125f;
    q.y *= 0.125f;
    qf[dk] = q;
  }

  v8f o[4];
  float Mr[8], Sr[8];
#pragma unroll
  for (int t = 0; t < 4; ++t) o[t] = (v8f){0, 0, 0, 0, 0, 0, 0, 0};
#pragma unroll
  for (int r = 0; r < 8; ++r) { Mr[r] = -1e30f; Sr[r] = 0.f; }

  for (int kb = 0; kb <= qtile; ++kb) {
    const int kbase = kb * 16;
    const float* krow = qkv + ((size_t)(b * S + kbase + lr)) * D3 + D + h * HD + lh * 2;
    v8f s = {0, 0, 0, 0, 0, 0, 0, 0};
#pragma unroll
    for (int dk = 0; dk < 16; ++dk) {
      v2f kv = *(const v2f*)(krow + dk * 4);
      s = WMMA_F32(qf[dk], kv, s);
    }
    if (kb == qtile) {  // diagonal block: mask key(n=lr) > query(m=lh*8+r)
#pragma unroll
      for (int r = 0; r < 8; ++r)
        if (lr > lh * 8 + r) s[r] = -1e30f;
    }
    float rm[8], p[8], rs[8], sc[8];
#pragma unroll
    for (int r = 0; r < 8; ++r) rm[r] = s[r];
#pragma unroll
    for (int off = 1; off < 16; off <<= 1)
#pragma unroll
      for (int r = 0; r < 8; ++r) rm[r] = fmaxf(rm[r], __shfl_xor(rm[r], off, 32));
#pragma unroll
    for (int r = 0; r < 8; ++r) {
      float nm = fmaxf(Mr[r], rm[r]);
      sc[r] = __expf(Mr[r] - nm);
      Mr[r] = nm;
      p[r] = __expf(s[r] - nm);
      rs[r] = p[r];
    }
#pragma unroll
    for (int off = 1; off < 16; off <<= 1)
#pragma unroll
      for (int r = 0; r < 8; ++r) rs[r] += __shfl_xor(rs[r], off, 32);
#pragma unroll
    for (int r = 0; r < 8; ++r) Sr[r] = Sr[r] * sc[r] + rs[r];
#pragma unroll
    for (int t = 0; t < 4; ++t)
#pragma unroll
      for (int r = 0; r < 8; ++r) o[t][r] *= sc[r];

    // P (C-layout [query, key]) -> LDS -> A-fragments
#pragma unroll
    for (int r = 0; r < 8; ++r) wp[(lh * 8 + r) * 18 + lr] = p[r];
    v2f af[4];
#pragma unroll
    for (int kc = 0; kc < 4; ++kc)
      af[kc] = *(const v2f*)(wp + lr * 18 + kc * 4 + lh * 2);

    const float* vb = qkv + ((size_t)(b * S + kbase + lh * 2)) * D3 + 2 * D + h * HD + lr;
#pragma unroll
    for (int kc = 0; kc < 4; ++kc) {
      const float* v0 = vb + (size_t)(kc * 4) * D3;
#pragma unroll
      for (int t = 0; t < 4; ++t) {
        v2f bv;
        bv.x = v0[t * 16];
        bv.y = v0[(size_t)D3 + t * 16];
        o[t] = WMMA_F32(af[kc], bv, o[t]);
      }
    }
  }

  float inv[8];
#pragma unroll
  for (int r = 0; r < 8; ++r) inv[r] = 1.0f / Sr[r];
#pragma unroll
  for (int t = 0; t < 4; ++t)
#pragma unroll
    for (int r = 0; r < 8; ++r)
      ao[((size_t)(b * S + qbase + lh * 8 + r)) * D + h * HD + t * 16 + lr] = o[t][r] * inv[r];
}

// ---------------- final: out = y1 + ffn + mhc(x_delta, g, H) ----------------
__global__ __launch_bounds__(256) void k_final(const float* __restrict__ y1,
                                               const float* __restrict__ ffn,
                                               const float* __restrict__ xd,
                                               const float* __restrict__ gres,
                                               const float* __restrict__ Hm,
                                               float* __restrict__ out) {
  __shared__ float gh[B][NS][NS];
  if (threadIdx.x < B * NS * NS) {
    int b = threadIdx.x / (NS * NS);
    int m = (threadIdx.x / NS) % NS;
    int n = threadIdx.x % NS;
    gh[b][m][n] = Hm[m * NS + n] * gres[b * NS + n];
  }
  __syncthreads();
  size_t idx = (size_t)blockIdx.x * 256 + threadIdx.x;
  int dcol = (int)(idx % D);
  size_t bs = idx / D;
  int b = (int)(bs / S);
  int m = dcol >> 7, dp = dcol & (SDIM - 1);
  const float* xr = xd + bs * D + dp;
  float a = y1[idx] + ffn[idx];
#pragma unroll
  for (int n = 0; n < NS; ++n) a += gh[b][m][n] * xr[n * SDIM];
  out[idx] = a;
}

extern "C" void kernel_launch(void* const* d_in, const int* in_sizes, int n_in,
                              void* d_out, int out_size, void* d_ws, size_t ws_size,
                              hipStream_t stream) {
  (void)in_sizes; (void)n_in; (void)out_size; (void)ws_size;
  const float* x     = (const float*)d_in[0];
  const float* decay = (const float*)d_in[3];
  const float* gatew = (const float*)d_in[4];
  const float* gateb = (const float*)d_in[5];
  const float* phi   = (const float*)d_in[6];
  const float* ln1   = (const float*)d_in[7];
  const float* ln2   = (const float*)d_in[8];
  const float* w1    = (const float*)d_in[9];
  const float* b1    = (const float*)d_in[10];
  const float* w2    = (const float*)d_in[11];
  const float* b2    = (const float*)d_in[12];
  const float* ipw   = (const float*)d_in[13];
  const float* ipb   = (const float*)d_in[14];
  const float* ow    = (const float*)d_in[15];
  const float* ob    = (const float*)d_in[16];
  float* ws = (float*)d_ws;
  float* xd  = ws + OFF_XD;
  float* nb  = ws + OFF_N;
  float* big = ws + OFF_BIG;
  float* ao  = ws + OFF_AO;
  float* y1  = ws + OFF_Y1;
  float* ctr = ws + OFF_CTR;
  float* sib = ws + OFF_SIN;
  float* ps  = ws + OFF_PS;
  float* xst = ws + OFF_XST;
  float* gr  = ws + OFF_GR;
  float* Hm  = ws + OFF_H;
  float* out = (float*)d_out;

  // 1) delta operator (chunked scan) + x_state partials
  k_delta_a<<<B * D * CH / 256, 256, 0, stream>>>(x, decay, ctr);
  k_delta_b<<<(B * D + 255) / 256, 256, 0, stream>>>(ctr, decay, sib);
  k_delta_c<<<B * D * CH / 256, 256, 0, stream>>>(x, decay, sib, xd, ps);
  // 2) x_state + gates + sinkhorn
  k_combo<<<1, 256, 0, stream>>>(ps, gatew, gateb, phi, xst, gr, Hm);
  // 3) normed1 = rms(x_delta)
  k_rms<<<ROWS, 256, 0, stream>>>(xd, ln1, nb);
  // 4) qkv = normed1 @ in_proj^T + b
  k_gemm<0, false><<<dim3(ROWS / 64, D3 / 128), 256, 0, stream>>>(nb, ipw, ipb, nullptr, big,
                                                                  ROWS, D3, D);
  // 5) causal flash attention -> attn_o
  k_flash<<<dim3(S / 16 / 4, B * NH), 128, 0, stream>>>(big, ao);
  // 6) y1 = x_delta + attn_o @ out_w^T + out_b
  k_gemm<0, true><<<dim3(ROWS / 64, D / 128), 256, 0, stream>>>(ao, ow, ob, xd, y1,
                                                                ROWS, D, D);
  // 7) normed2 = rms(y1)
  k_rms<<<ROWS, 256, 0, stream>>>(y1, ln2, nb);
  // 8) MLP in two row-halves (hidden reuses qkv buffer); ffn overwrites attn_o buffer
  for (int half = 0; half < 2; ++half) {
    const float* a2 = nb + (size_t)half * (ROWS / 2) * D;
    float* ffnh = ao + (size_t)half * (ROWS / 2) * D;
    k_gemm<1, false><<<dim3(ROWS / 2 / 64, D4 / 128), 256, 0, stream>>>(a2, w1, b1, nullptr,
                                                                        big, ROWS / 2, D4, D);
    k_gemm<0, false><<<dim3(ROWS / 2 / 64, D / 128), 256, 0, stream>>>(big, w2, b2, nullptr,
                                                                       ffnh, ROWS / 2, D, D4);
  }
  // 9) out = y1 + ffn + mhc
  k_final<<<(unsigned)((size_t)ROWS * D / 256), 256, 0, stream>>>(y1, ao, xd, gr, Hm, out);
}